// MinibatchDiscrimination_12962211299446
// MI455X (gfx1250) — compile-verified
//
#include <hip/hip_runtime.h>

typedef __attribute__((ext_vector_type(2))) float v2f;
typedef __attribute__((ext_vector_type(8))) float v8f;
typedef __attribute__((ext_vector_type(4))) int   v4i;

#define N_ROWS 1024
#define IN_F   1024
#define OUT_F  128
#define KDIM   16
#define COLS   (OUT_F * KDIM)      // 2048
#define OUT_STRIDE (IN_F + OUT_F)  // 1152

// LDS layout for mbd_main: M strip [16][MROW] + sq [1024]
// MROW mod 64 == 8 -> K-rows (k, k+2) hit disjoint bank sets for the two
// half-waves; rows stay 16B aligned (1032*4 = 4128 = 16*258).
#define MROW   1032
#define SQ_OFF (16 * MROW)
#define LDS_FLOATS (SQ_OFF + 1024)

// ---------------------------------------------------------------------------
// CDNA5 async global->LDS 16B copy (ASYNCcnt-tracked) + wait helpers
// Builtin signature (from clang diagnostic): (v4i AS1*, v4i AS3*, imm, imm)
// ---------------------------------------------------------------------------
__device__ __forceinline__ void async_copy16(const float* gp, float* lp) {
#if __has_builtin(__builtin_amdgcn_global_load_async_to_lds_b128)
    __builtin_amdgcn_global_load_async_to_lds_b128(
        (__attribute__((address_space(1))) v4i*)(void*)gp,
        (__attribute__((address_space(3))) v4i*)(void*)lp, 0, 0);
#else
    unsigned lds_addr = (unsigned)(unsigned long long)(const void*)lp;
    asm volatile("global_load_async_to_lds_b128 %0, %1, off"
                 :: "v"(lds_addr), "v"(gp) : "memory");
#endif
}

__device__ __forceinline__ void wait_async0() {
#if __has_builtin(__builtin_amdgcn_s_wait_asynccnt)
    __builtin_amdgcn_s_wait_asynccnt(0);
#else
    asm volatile("s_wait_asynccnt 0x0" ::: "memory");
#endif
}

// ---------------------------------------------------------------------------
// Kernel 1: copy x into out[:, 0:1024]  (float4 vectorized; 1152 % 4 == 0)
// ---------------------------------------------------------------------------
__global__ void mbd_copy_x(const float* __restrict__ x, float* __restrict__ out) {
    int idx = blockIdx.x * blockDim.x + threadIdx.x;  // one float4 each
    int i  = idx >> 8;                                // 256 float4 per row
    int c4 = idx & 255;
    const float4* src = (const float4*)x;
    float4*       dst = (float4*)out;
    dst[i * (OUT_STRIDE / 4) + c4] = src[i * (IN_F / 4) + c4];
}

// ---------------------------------------------------------------------------
// Kernel 2: Mt = (x @ T)^T stored as [col=2048][row=1024] f32 in workspace.
// One wave per 16x16 output tile; K-loop = 256 chained V_WMMA_F32_16X16X4_F32.
// ---------------------------------------------------------------------------
__global__ void mbd_gemm(const float* __restrict__ x, const float* __restrict__ T,
                         float* __restrict__ Mt) {
    int wave = threadIdx.x >> 5;
    int lane = threadIdx.x & 31;
    int t  = blockIdx.x * (blockDim.x >> 5) + wave;
    int it = t >> 7;                 // i-tile 0..63
    int ct = t & 127;                // col-tile 0..127
    int i0 = it * 16, c0 = ct * 16;
    int r  = lane & 15;
    int h  = lane >> 4;              // half-wave select (K offset +2)

    v8f acc = {};
    for (int k = 0; k < IN_F; k += 4) {
        v2f a, b;
        a.x = x[(i0 + r) * IN_F + k + 2 * h];
        a.y = x[(i0 + r) * IN_F + k + 2 * h + 1];
        b.x = T[(k + 2 * h) * COLS + c0 + r];
        b.y = T[(k + 2 * h + 1) * COLS + c0 + r];
        acc = __builtin_amdgcn_wmma_f32_16x16x4_f32(
            false, a, false, b, (short)0, acc, false, false);
    }
    // D layout: lane holds col n=r, rows m = v + 8*h. Store transposed.
#pragma unroll
    for (int v = 0; v < 8; ++v) {
        Mt[(c0 + r) * N_ROWS + (i0 + v + 8 * h)] = acc[v];
    }
}

// ---------------------------------------------------------------------------
// Kernel 3: sq[b][i] = sum_k Mt[b*16+k][i]^2   ([B][N] layout, lane-contiguous)
// ---------------------------------------------------------------------------
__global__ void mbd_sq(const float* __restrict__ Mt, float* __restrict__ sqv) {
    int idx = blockIdx.x * blockDim.x + threadIdx.x;  // b*1024 + i
    int b = idx >> 10;
    int i = idx & 1023;
    float s = 0.f;
#pragma unroll
    for (int k = 0; k < KDIM; ++k) {
        float m = Mt[(b * KDIM + k) * N_ROWS + i];
        s += m * m;
    }
    sqv[b * N_ROWS + i] = s;
}

// ---------------------------------------------------------------------------
// Kernel 4: block = 8 waves sharing feature b. Stage Mb [16][1024] (64KB) and
// sq_b (4KB) into LDS via async global->LDS b128, then each wave handles one
// 16-row i-tile: loop j-tiles, 4x WMMA f32 16x16x4 from ds_load operands,
// fuse exp(min(2G - sq_i - sq_j, 0)) on accumulator regs, butterfly-reduce.
// ---------------------------------------------------------------------------
__global__ void mbd_main(const float* __restrict__ Mt, const float* __restrict__ sqv,
                         float* __restrict__ out) {
    __shared__ float smem[LDS_FLOATS];

    int wave = threadIdx.x >> 5;     // 0..7
    int lane = threadIdx.x & 31;
    int b    = blockIdx.x >> 3;      // feature 0..127
    int grp  = blockIdx.x & 7;       // i-tile group 0..7
    int it   = grp * 8 + wave;       // i-tile 0..63
    int i0   = it * 16;
    int r    = lane & 15;
    int h    = lane >> 4;

    const float* gM  = Mt  + b * KDIM * N_ROWS;   // [16][1024]
    const float* gSq = sqv + b * N_ROWS;

    // ---- Stage Mb + sq_b into LDS (async, 16B per lane per op) ----
    for (int c = threadIdx.x; c < 16 * 256; c += 256) {   // 4096 chunks of 16B
        int kr = c >> 8;          // row 0..15
        int cc = c & 255;         // 16B chunk within row
        async_copy16(gM + kr * N_ROWS + cc * 4, &smem[kr * MROW + cc * 4]);
    }
    async_copy16(gSq + threadIdx.x * 4, &smem[SQ_OFF + threadIdx.x * 4]);
    wait_async0();
    __syncthreads();

    const float* lM  = smem;              // [16][MROW]
    const float* lSq = smem + SQ_OFF;     // [1024]

    // A-tile for rows i0..i0+15, K=16 (4 chunks of v2f) from LDS
    v2f A[4];
#pragma unroll
    for (int kk = 0; kk < 4; ++kk) {
        A[kk].x = lM[(4 * kk + 2 * h) * MROW + i0 + r];
        A[kk].y = lM[(4 * kk + 2 * h + 1) * MROW + i0 + r];
    }
    float sqi[8];
#pragma unroll
    for (int v = 0; v < 8; ++v) sqi[v] = lSq[i0 + v + 8 * h];

    float racc[8] = {};
    for (int j0 = 0; j0 < N_ROWS; j0 += 16) {
        float sqj = lSq[j0 + r];     // this lane's column j for every VGPR row
        v8f c = {};
#pragma unroll
        for (int kk = 0; kk < 4; ++kk) {
            v2f B;
            B.x = lM[(4 * kk + 2 * h) * MROW + j0 + r];
            B.y = lM[(4 * kk + 2 * h + 1) * MROW + j0 + r];
            c = __builtin_amdgcn_wmma_f32_16x16x4_f32(
                false, A[kk], false, B, (short)0, c, false, false);
        }
#pragma unroll
        for (int v = 0; v < 8; ++v) {
            // exp(-max(d2,0)) = exp(min(2G - sqi - sqj, 0))
            float e = fminf(2.0f * c[v] - sqi[v] - sqj, 0.0f);
            racc[v] += __expf(e);
        }
    }
    // Reduce over the 16 lanes of each half-wave (columns j within group)
#pragma unroll
    for (int v = 0; v < 8; ++v) {
        float s = racc[v];
#pragma unroll
        for (int m = 1; m < 16; m <<= 1) s += __shfl_xor(s, m, 32);
        racc[v] = s;
    }
    if (r == 0) {
#pragma unroll
        for (int v = 0; v < 8; ++v) {
            int row = i0 + v + 8 * h;
            out[row * OUT_STRIDE + IN_F + b] = racc[v] - 1.0f;  // minus self-term
        }
    }
}

// ---------------------------------------------------------------------------
extern "C" void kernel_launch(void* const* d_in, const int* in_sizes, int n_in,
                              void* d_out, int out_size, void* d_ws, size_t ws_size,
                              hipStream_t stream) {
    const float* x = (const float*)d_in[0];   // [1024][1024]
    const float* T = (const float*)d_in[1];   // [1024][2048]
    float* out = (float*)d_out;               // [1024][1152]

    float* Mt  = (float*)d_ws;                // [2048][1024]  (8 MB)
    float* sqv = Mt + (size_t)COLS * N_ROWS;  // [128][1024]   (0.5 MB)

    // 1) out[:, :1024] = x
    mbd_copy_x<<<(N_ROWS * (IN_F / 4)) / 256, 256, 0, stream>>>(x, out);
    // 2) Mt = (x @ T)^T   — 64 x 128 tiles, 8 waves/block
    mbd_gemm<<<(64 * 128) / 8, 256, 0, stream>>>(x, T, Mt);
    // 3) sq
    mbd_sq<<<(N_ROWS * OUT_F) / 256, 256, 0, stream>>>(Mt, sqv);
    // 4) fused Gram + exp + row-reduce -> out[:, 1024:]
    //    128 features x 8 i-tile groups, 8 waves/block share one feature's LDS
    mbd_main<<<128 * 8, 256, 0, stream>>>(Mt, sqv, out);
}